// GlobalModel_18159121728221
// MI455X (gfx1250) — compile-verified
//
#include <hip/hip_runtime.h>
#include <hip/hip_bf16.h>

#define N_NODES   50000
#define N_EDGES   1600000
#define N_EDGE_F  128
#define GLOBAL_F  64
#define N_GRAPHS  512
#define HIDDEN    64
#define IN1       (N_EDGE_F + GLOBAL_F)   // 192

typedef __attribute__((ext_vector_type(2))) float v2f;
typedef __attribute__((ext_vector_type(8))) float v8f;

// Workspace layout (floats): [0, 512*128) = per-graph feature sums,
//                            [512*128, +512) = per-graph edge counts
#define WS_SUMS  0
#define WS_CNT   (N_GRAPHS * N_EDGE_F)
#define WS_TOTAL (WS_CNT + N_GRAPHS)

// ---------------------------------------------------------------------------
// Kernel 1: zero the scatter accumulators (must run every call; the harness
// does not re-initialize d_ws between replays).
// ---------------------------------------------------------------------------
__global__ void zero_ws_kernel(float* __restrict__ ws) {
    int i = blockIdx.x * blockDim.x + threadIdx.x;
    if (i < WS_TOTAL) ws[i] = 0.0f;
}

// ---------------------------------------------------------------------------
// Kernel 2: scatter-add edge_attr into per-graph sums, privatized in LDS.
// Feature dim split into two 64-wide phases so the LDS accumulator is
// 512*64*4 = 128KB (+2KB counts), safely inside the WGP's 320KB LDS.
// One edge per wave iteration: 32 lanes x float2 covers 64 features,
// coalesced 256B read; ds_add_f32 into LDS (bank-conflict free), then a
// single global_atomic_add_f32 flush per phase.
// ---------------------------------------------------------------------------
__global__ __launch_bounds__(256) void scatter_kernel(
    const int*   __restrict__ edge_index,   // [2, N_EDGES], row 0 = src node
    const int*   __restrict__ batch,        // [N_NODES] graph id (sorted)
    const float* __restrict__ edge_attr,    // [N_EDGES, 128]
    float*       __restrict__ ws) {
    __shared__ float s_acc[N_GRAPHS * 64];  // 128 KB
    __shared__ float s_cnt[N_GRAPHS];       // 2 KB
    float* sums = ws + WS_SUMS;
    float* cnts = ws + WS_CNT;

    const int tid    = threadIdx.x;
    const int lane   = tid & 31;
    const int wave   = blockIdx.x * (256 / 32) + (tid >> 5);
    const int nwaves = gridDim.x * (256 / 32);

    for (int ph = 0; ph < 2; ++ph) {
        for (int i = tid; i < N_GRAPHS * 64; i += 256) s_acc[i] = 0.0f;
        if (ph == 0)
            for (int i = tid; i < N_GRAPHS; i += 256) s_cnt[i] = 0.0f;
        __syncthreads();

        for (int e = wave; e < N_EDGES; e += nwaves) {
            int src = edge_index[e];        // edge_index[0][e]
            int g   = batch[src];
            const float2 v = *(const float2*)(edge_attr +
                               (size_t)e * N_EDGE_F + ph * 64 + lane * 2);
            atomicAdd(&s_acc[g * 64 + lane * 2 + 0], v.x);   // ds_add_f32
            atomicAdd(&s_acc[g * 64 + lane * 2 + 1], v.y);   // ds_add_f32
            if (ph == 0 && lane == 0) atomicAdd(&s_cnt[g], 1.0f);
        }
        __syncthreads();

        for (int i = tid; i < N_GRAPHS * 64; i += 256) {
            int g = i >> 6, f = i & 63;
            atomicAdd(&sums[g * N_EDGE_F + ph * 64 + f], s_acc[i]);
        }
        if (ph == 0)
            for (int i = tid; i < N_GRAPHS; i += 256)
                atomicAdd(&cnts[i], s_cnt[i]);
        __syncthreads();
    }
}

// ---------------------------------------------------------------------------
// Kernel 3: mean + concat + 2-layer MLP via V_WMMA_F32_16X16X4_F32.
// 32 blocks (one per 16-graph row tile), 4 waves per block (one per 16-col
// N-tile of the 64-wide hidden/output dims).
//
// f32 WMMA fragment layouts (ISA 7.12.2):
//   A 16x4 : lane[3:0]=M row, lane[4]=K-pair select; VGPR0 = K even, VGPR1 = K odd
//   B 4x16 : lane[3:0]=N col, lane[4]=K-pair select; VGPR0 = K even, VGPR1 = K odd
//   C/D    : VGPR i -> M = i + 8*lane[4], N = lane[3:0]
// ---------------------------------------------------------------------------
__global__ __launch_bounds__(128) void mlp_kernel(
    const float* __restrict__ u,    // [512, 64]
    const float* __restrict__ ws,   // sums + counts
    const float* __restrict__ W1,   // [192, 64]
    const float* __restrict__ b1,   // [64]
    const float* __restrict__ W2,   // [64, 64]
    const float* __restrict__ b2,   // [64]
    float*       __restrict__ out)  // [512, 64]
{
    __shared__ float sA[16][IN1 + 4];      // +4 pad: LDS bank stride 196 % 64 = 4
    __shared__ float sH[16][HIDDEN + 4];   // +4 pad
    const float* sums = ws + WS_SUMS;
    const float* cnts = ws + WS_CNT;

    const int tid = threadIdx.x;
    const int g0  = blockIdx.x * 16;

    // Stage [u | mean] tile: 16 rows x 192 cols
    for (int i = tid; i < 16 * IN1; i += 128) {
        int r = i / IN1, c = i - r * IN1;
        float v;
        if (c < GLOBAL_F) {
            v = u[(g0 + r) * GLOBAL_F + c];
        } else {
            float cnt = cnts[g0 + r];
            v = sums[(g0 + r) * N_EDGE_F + (c - GLOBAL_F)] / fmaxf(cnt, 1.0f);
        }
        sA[r][c] = v;
    }
    __syncthreads();

    const int wave = tid >> 5;
    const int lane = tid & 31;
    const int n0   = wave * 16;      // this wave's N-tile
    const int row  = lane & 15;      // M row (A) / N col (B)
    const int koff = lane >> 4;      // K-pair select

    // ---- GEMM1: h = relu([u|mean] @ W1 + b1), K = 192 ----
    v8f acc = {};
    for (int k = 0; k < IN1; k += 4) {
        int ka = k + 2 * koff;
        v2f a, b;
        a.x = sA[row][ka];
        a.y = sA[row][ka + 1];
        b.x = W1[ka * HIDDEN + n0 + row];
        b.y = W1[(ka + 1) * HIDDEN + n0 + row];
        acc = __builtin_amdgcn_wmma_f32_16x16x4_f32(
                  false, a, false, b, (short)0, acc, false, false);
    }
#pragma unroll
    for (int i = 0; i < 8; ++i) {
        int M = i + 8 * koff;
        int N = n0 + row;
        sH[M][N] = fmaxf(acc[i] + b1[N], 0.0f);
    }
    __syncthreads();

    // ---- GEMM2: out = h @ W2 + b2, K = 64 ----
    v8f acc2 = {};
    for (int k = 0; k < HIDDEN; k += 4) {
        int ka = k + 2 * koff;
        v2f a, b;
        a.x = sH[row][ka];
        a.y = sH[row][ka + 1];
        b.x = W2[ka * GLOBAL_F + n0 + row];
        b.y = W2[(ka + 1) * GLOBAL_F + n0 + row];
        acc2 = __builtin_amdgcn_wmma_f32_16x16x4_f32(
                   false, a, false, b, (short)0, acc2, false, false);
    }
#pragma unroll
    for (int i = 0; i < 8; ++i) {
        int M = i + 8 * koff;
        int N = n0 + row;
        out[(g0 + M) * GLOBAL_F + N] = acc2[i] + b2[N];
    }
}

// ---------------------------------------------------------------------------
extern "C" void kernel_launch(void* const* d_in, const int* in_sizes, int n_in,
                              void* d_out, int out_size, void* d_ws, size_t ws_size,
                              hipStream_t stream) {
    (void)in_sizes; (void)n_in; (void)out_size; (void)ws_size;
    // setup_inputs order: x, edge_index, edge_attr, u, batch, W1, b1, W2, b2
    const int*   edge_index = (const int*)  d_in[1];
    const float* edge_attr  = (const float*)d_in[2];
    const float* u          = (const float*)d_in[3];
    const int*   batch      = (const int*)  d_in[4];
    const float* W1         = (const float*)d_in[5];
    const float* b1         = (const float*)d_in[6];
    const float* W2         = (const float*)d_in[7];
    const float* b2         = (const float*)d_in[8];
    float* out = (float*)d_out;
    float* ws  = (float*)d_ws;

    zero_ws_kernel<<<(WS_TOTAL + 255) / 256, 256, 0, stream>>>(ws);
    scatter_kernel<<<128, 256, 0, stream>>>(edge_index, batch, edge_attr, ws);
    mlp_kernel<<<N_GRAPHS / 16, 128, 0, stream>>>(u, ws, W1, b1, W2, b2, out);
}